// HyperKA_3212635538064
// MI455X (gfx1250) — compile-verified
//
#include <hip/hip_runtime.h>
#include <math.h>

#define EPS       1e-6f
#define PROJ_EPS  1e-5f
#define DD        64
#define TILE_ROWS 32
#define TPITCH    65   // padded LDS pitch: (m*65+k)%64 distinct per m -> no bank conflicts

typedef float v2f __attribute__((ext_vector_type(2)));
typedef float v8f __attribute__((ext_vector_type(8)));

// ---------------- wave32 helpers ----------------
__device__ __forceinline__ float wave_sum(float v) {
    v += __shfl_xor(v, 1, 32);
    v += __shfl_xor(v, 2, 32);
    v += __shfl_xor(v, 4, 32);
    v += __shfl_xor(v, 8, 32);
    v += __shfl_xor(v, 16, 32);
    return v;
}
__device__ __forceinline__ float artanh_f(float z) {
    return 0.5f * logf((1.0f + z) / (1.0f - z));
}

// ---------------- kernel 1: fused (hyp_proj)+log_map_zero + GEMM via fp32 WMMA ----
// h[rb..rb+31][0..63] = log_map_zero( (from_x ? hyp_proj(in) : in) ) @ W
__global__ __launch_bounds__(256)
void hgcn_transform_wmma(const float* __restrict__ in, const float* __restrict__ W,
                         float* __restrict__ h, int N, int from_x) {
    __shared__ float ldsW[DD * DD];
    __shared__ float ldsT[TILE_ROWS * TPITCH];
    __shared__ float ldsS[TILE_ROWS];

    const int tid = threadIdx.x;
    const int rb  = blockIdx.x * TILE_ROWS;

    // stage W (64x64, row-major)
    #pragma unroll
    for (int i = 0; i < (DD * DD) / 256; ++i)
        ldsW[i * 256 + tid] = W[i * 256 + tid];

    // stage 32 input rows, zero-padded past N
    #pragma unroll
    for (int i = 0; i < (TILE_ROWS * DD) / 256; ++i) {
        int e = i * 256 + tid;
        int r = e >> 6, c = e & 63;
        float v = (rb + r < N) ? in[(size_t)(rb + r) * DD + c] : 0.0f;
        ldsT[r * TPITCH + c] = v;
    }
    __syncthreads();

    // per-row combined scale: optional ball projection, then log map
    if (tid < TILE_ROWS) {
        float s = 0.0f;
        #pragma unroll
        for (int c = 0; c < DD; ++c) { float v = ldsT[tid * TPITCH + c]; s += v * v; }
        float n  = sqrtf(fmaxf(s, EPS * EPS));
        float sp = 1.0f;
        if (from_x) { sp = fminf((1.0f - PROJ_EPS) / n, 1.0f); n *= sp; }
        float sl = artanh_f(fminf(n, 1.0f - PROJ_EPS)) / n;
        ldsS[tid] = sp * sl;
    }
    __syncthreads();

    // 8 waves: (row-tile 0..1) x (col-tile 0..3), each a 16x16 C tile, K=64 in 16 steps of 4
    const int wv   = tid >> 5;
    const int lane = tid & 31;
    const int rt   = wv >> 2;
    const int ct   = wv & 3;
    const int ml   = rt * 16 + (lane & 15);       // A row (M)
    const int nn   = ct * 16 + (lane & 15);       // C column (N)
    const int koff = (lane < 16) ? 0 : 2;         // ISA: lanes 16-31 hold K+2
    const float sc = ldsS[ml];

    v8f acc = {0.f, 0.f, 0.f, 0.f, 0.f, 0.f, 0.f, 0.f};
    #pragma unroll
    for (int kk = 0; kk < 16; ++kk) {
        const int kb = kk * 4 + koff;
        v2f a, b;
        a.x = ldsT[ml * TPITCH + kb]     * sc;   // A[m][kb],  A[m][kb+1]
        a.y = ldsT[ml * TPITCH + kb + 1] * sc;
        b.x = ldsW[kb * DD + nn];                // B[kb][n],  B[kb+1][n]
        b.y = ldsW[(kb + 1) * DD + nn];
        acc = __builtin_amdgcn_wmma_f32_16x16x4_f32(
                  false, a, false, b, (short)0, acc, false, false);
    }

    // C/D layout: VGPR r -> rows r (lanes 0-15) and r+8 (lanes 16-31)
    const int rhalf = (lane < 16) ? 0 : 8;
    #pragma unroll
    for (int r = 0; r < 8; ++r) {
        int grow = rb + rt * 16 + rhalf + r;
        if (grow < N) h[(size_t)grow * DD + nn] = acc[r];
    }
}

// ---------------- kernel 2: zero agg ----------------
__global__ void zero_f32(float* __restrict__ p, size_t n) {
    size_t i      = (size_t)blockIdx.x * blockDim.x + threadIdx.x;
    size_t stride = (size_t)gridDim.x * blockDim.x;
    for (; i < n; i += stride) p[i] = 0.0f;
}

// ---------------- kernel 3: edge scatter: agg[r] += val * h[c] ----------------
// one wave per edge; float2 per lane -> coalesced 256B gather, L2-resident atomics
__global__ __launch_bounds__(256)
void hgcn_scatter(const int* __restrict__ rows, const int* __restrict__ cols,
                  const float* __restrict__ vals, const float* __restrict__ h,
                  float* __restrict__ agg, int E) {
    int e    = (int)(((size_t)blockIdx.x * blockDim.x + threadIdx.x) >> 5);
    int lane = threadIdx.x & 31;
    if (e >= E) return;
    int   r = rows[e];
    int   c = cols[e];
    float v = vals[e];
    float2 hv = ((const float2*)h)[(size_t)c * 32 + lane];
    atomicAdd(&agg[(size_t)r * DD + lane * 2    ], v * hv.x);
    atomicAdd(&agg[(size_t)r * DD + lane * 2 + 1], v * hv.y);
}

// ---------------- kernel 4: per-node exp/proj/mobius-bias/(act)/mobius-residual ----
// one wave per node, 2 dims per lane, reductions via shfl_xor
__global__ __launch_bounds__(256)
void hgcn_post(const float* __restrict__ agg, const float* __restrict__ bias,
               const float* __restrict__ res, float* __restrict__ out,
               int N, int use_act, int res_is_x) {
    int node = blockIdx.x * 8 + (threadIdx.x >> 5);
    int lane = threadIdx.x & 31;
    if (node >= N) return;                  // uniform per-wave exit
    const int d0 = lane * 2;

    float ox = agg[(size_t)node * DD + d0];
    float oy = agg[(size_t)node * DD + d0 + 1];

    // exp_map_zero
    float n = sqrtf(fmaxf(wave_sum(ox * ox + oy * oy), EPS * EPS));
    float s = tanhf(n) / n;  ox *= s;  oy *= s;
    // hyp_proj
    n = sqrtf(fmaxf(wave_sum(ox * ox + oy * oy), EPS * EPS));
    s = fminf((1.0f - PROJ_EPS) / n, 1.0f);  ox *= s;  oy *= s;

    // bh = hyp_proj(exp_map_zero(b))
    float bx = bias[d0], by = bias[d0 + 1];
    n = sqrtf(fmaxf(wave_sum(bx * bx + by * by), EPS * EPS));
    s = tanhf(n) / n;  bx *= s;  by *= s;
    n = sqrtf(fmaxf(wave_sum(bx * bx + by * by), EPS * EPS));
    s = fminf((1.0f - PROJ_EPS) / n, 1.0f);  bx *= s;  by *= s;

    // out = hyp_proj(mobius_add(out, bh))
    {
        float x2 = wave_sum(ox * ox + oy * oy);
        float y2 = wave_sum(bx * bx + by * by);
        float xy = wave_sum(ox * bx + oy * by);
        float den = fmaxf(1.0f + 2.0f * xy + x2 * y2, EPS);
        float cx = 1.0f + 2.0f * xy + y2, cy = 1.0f - x2;
        float nx = cx * ox + cy * bx, ny = cx * oy + cy * by;
        ox = nx / den;  oy = ny / den;
    }
    n = sqrtf(fmaxf(wave_sum(ox * ox + oy * oy), EPS * EPS));
    s = fminf((1.0f - PROJ_EPS) / n, 1.0f);  ox *= s;  oy *= s;

    if (use_act) {
        // hyp_proj(exp_map_zero(tanh(log_map_zero(out))))
        n = sqrtf(fmaxf(wave_sum(ox * ox + oy * oy), EPS * EPS));
        s = artanh_f(fminf(n, 1.0f - PROJ_EPS)) / n;
        float tx = tanhf(ox * s), ty = tanhf(oy * s);
        n = sqrtf(fmaxf(wave_sum(tx * tx + ty * ty), EPS * EPS));
        s = tanhf(n) / n;  ox = tx * s;  oy = ty * s;
        n = sqrtf(fmaxf(wave_sum(ox * ox + oy * oy), EPS * EPS));
        s = fminf((1.0f - PROJ_EPS) / n, 1.0f);  ox *= s;  oy *= s;
    }

    // residual: r = res_is_x ? hyp_proj(x_row) : res_row
    float rx = res[(size_t)node * DD + d0], ry = res[(size_t)node * DD + d0 + 1];
    if (res_is_x) {
        n = sqrtf(fmaxf(wave_sum(rx * rx + ry * ry), EPS * EPS));
        s = fminf((1.0f - PROJ_EPS) / n, 1.0f);  rx *= s;  ry *= s;
    }
    // out = hyp_proj(mobius_add(out, r))
    {
        float x2 = wave_sum(ox * ox + oy * oy);
        float y2 = wave_sum(rx * rx + ry * ry);
        float xy = wave_sum(ox * rx + oy * ry);
        float den = fmaxf(1.0f + 2.0f * xy + x2 * y2, EPS);
        float cx = 1.0f + 2.0f * xy + y2, cy = 1.0f - x2;
        float nx = cx * ox + cy * rx, ny = cx * oy + cy * ry;
        ox = nx / den;  oy = ny / den;
    }
    n = sqrtf(fmaxf(wave_sum(ox * ox + oy * oy), EPS * EPS));
    s = fminf((1.0f - PROJ_EPS) / n, 1.0f);  ox *= s;  oy *= s;

    out[(size_t)node * DD + d0]     = ox;
    out[(size_t)node * DD + d0 + 1] = oy;
}

// ---------------- launcher ----------------
extern "C" void kernel_launch(void* const* d_in, const int* in_sizes, int n_in,
                              void* d_out, int out_size, void* d_ws, size_t ws_size,
                              hipStream_t stream) {
    const float* x    = (const float*)d_in[0];
    const float* vals = (const float*)d_in[1];
    const float* W1   = (const float*)d_in[2];
    const float* b1   = (const float*)d_in[3];
    const float* W2   = (const float*)d_in[4];
    const float* b2   = (const float*)d_in[5];
    const int*   rows = (const int*)d_in[6];
    const int*   cols = (const int*)d_in[7];

    const int N = in_sizes[0] / DD;
    const int E = in_sizes[1];
    const size_t NB = (size_t)N * DD;

    float* h   = (float*)d_ws;     // N*64
    float* agg = h   + NB;         // N*64
    float* h1  = agg + NB;         // N*64
    float* out = (float*)d_out;

    dim3 blk(256);
    const int gGemm = (N + TILE_ROWS - 1) / TILE_ROWS;
    const int gScat = (E + 7) / 8;   // 8 edge-waves per block
    const int gPost = (N + 7) / 8;   // 8 node-waves per block

    // ---- layer 1 (input: hyp_proj(x), activation on, residual e0) ----
    hgcn_transform_wmma<<<gGemm, blk, 0, stream>>>(x, W1, h, N, /*from_x=*/1);
    zero_f32<<<2048, blk, 0, stream>>>(agg, NB);
    hgcn_scatter<<<gScat, blk, 0, stream>>>(rows, cols, vals, h, agg, E);
    hgcn_post<<<gPost, blk, 0, stream>>>(agg, b1, x, h1, N, /*use_act=*/1, /*res_is_x=*/1);

    // ---- layer 2 (input: h1, no activation, residual h1) ----
    hgcn_transform_wmma<<<gGemm, blk, 0, stream>>>(h1, W2, h, N, /*from_x=*/0);
    zero_f32<<<2048, blk, 0, stream>>>(agg, NB);
    hgcn_scatter<<<gScat, blk, 0, stream>>>(rows, cols, vals, h, agg, E);
    hgcn_post<<<gPost, blk, 0, stream>>>(agg, b2, h1, out, N, /*use_act=*/0, /*res_is_x=*/0);
}